// KDELayer_22832046146207
// MI455X (gfx1250) — compile-verified
//
#include <hip/hip_runtime.h>
#include <math.h>

// KDE layer: out[n, b*32+c] = (coef/R) * sum_r exp(-50*(x[n,r,c] - b/31)^2)
// Factorized:  exp(-50(x-v)^2) = exp(-50x^2) * t^b * exp(-50 v^2),  t = exp(100x/31)
// => 2 transcendentals (v_exp_f32) per input element + 32-step multiply chain,
//    p_b = exp(-50x^2 + 100xb/31) stays within [e^-50, e^+50] (fp32-safe).

#define KDE_C 32
#define KDE_B 32
#define KDE_R 20
#define SITES_PER_BLOCK 8
#define KDE_THREADS 256                                  // 8 sites * 32 channels
#define TILE_FLOATS (SITES_PER_BLOCK * KDE_R * KDE_C)    // 5120 floats = 20 KB

struct KdeScales { float s[KDE_B]; };                    // (coef/R)*exp(-50*(b/31)^2)

// Async global->LDS copy, NT hint (input streamed once; don't pollute L2).
// INST_OFFSET is added to BOTH the global address and the LDS address (ISA 15.18.3),
// so one base VGPR pair + one LDS-offset VGPR serve all five 4 KB chunks.
#define ASYNC_B128(lds0, ga0, imm)                                          \
    asm volatile("global_load_async_to_lds_b128 %0, %1, off offset:" #imm   \
                 " th:TH_LOAD_NT"                                           \
                 :: "v"(lds0), "v"(ga0) : "memory")

__global__ __launch_bounds__(KDE_THREADS)
void kde_kernel(const float* __restrict__ x, float* __restrict__ out, KdeScales tab)
{
    __shared__ float tile[TILE_FLOATS];
    const int tid = threadIdx.x;
    const float* gsrc = x + (size_t)blockIdx.x * TILE_FLOATS;

    // ---- CDNA5 async global->LDS staging: 5 x b128 per lane, 20 KB/block ----
    {
        const unsigned lds0 = (unsigned)(unsigned long long)&tile[tid * 4]; // addr[31:0] == LDS offset
        const unsigned long long ga0 = (unsigned long long)(const void*)(gsrc + tid * 4);
        ASYNC_B128(lds0, ga0, 0);
        ASYNC_B128(lds0, ga0, 4096);
        ASYNC_B128(lds0, ga0, 8192);
        ASYNC_B128(lds0, ga0, 12288);
        ASYNC_B128(lds0, ga0, 16384);
    }
#if __has_builtin(__builtin_amdgcn_s_wait_asynccnt)
    __builtin_amdgcn_s_wait_asynccnt(0);
#else
    asm volatile("s_wait_asynccnt 0x0" ::: "memory");
#endif
    __syncthreads();

    // Thread (s, c): one (site, channel) pair, 32 bin accumulators in VGPRs.
    const int c = tid & (KDE_C - 1);
    const int s = tid >> 5;

    const float LOG2E = 1.4426950408889634f;
    const float K_T   = (100.0f / 31.0f) * LOG2E;  // t = exp2(K_T * x)
    const float K_G   = -50.0f * LOG2E;            // g = exp2(K_G * x*x)

    float acc[KDE_B];
#pragma unroll
    for (int b = 0; b < KDE_B; ++b) acc[b] = 0.0f;

#pragma unroll
    for (int r = 0; r < KDE_R; ++r) {
        const float xv = tile[(s * KDE_R + r) * KDE_C + c];
        const float t  = __builtin_amdgcn_exp2f(xv * K_T);        // v_exp_f32
        float       p  = __builtin_amdgcn_exp2f(xv * xv * K_G);   // exp(-50 x^2)
#pragma unroll
        for (int b = 0; b < KDE_B; ++b) {   // add+mul independent per step -> VOPD dual-issue
            acc[b] += p;
            p *= t;
        }
    }

    // out[n, b*C + c]: 128B contiguous per wave-store, non-temporal (streamed once).
    const size_t n = (size_t)blockIdx.x * SITES_PER_BLOCK + s;
    float* orow = out + n * (size_t)(KDE_B * KDE_C) + c;
#pragma unroll
    for (int b = 0; b < KDE_B; ++b) {
        __builtin_nontemporal_store(acc[b] * tab.s[b], orow + (size_t)b * KDE_C);
    }
}

extern "C" void kernel_launch(void* const* d_in, const int* in_sizes, int n_in,
                              void* d_out, int out_size, void* d_ws, size_t ws_size,
                              hipStream_t stream)
{
    (void)n_in; (void)out_size; (void)d_ws; (void)ws_size;
    const float* x = (const float*)d_in[0];
    float* out = (float*)d_out;

    // Host-side constant table: (coef/R) * exp(-50 * (b/31)^2)
    KdeScales tab;
    const float coef = 3.9894228040143274f;     // 1/sqrt(2*pi*0.01)
    for (int b = 0; b < KDE_B; ++b) {
        const float v = (float)b / 31.0f;
        tab.s[b] = (coef / (float)KDE_R) * expf(-50.0f * v * v);
    }

    const int n_sites = in_sizes[0] / (KDE_R * KDE_C);   // 131072
    dim3 grid(n_sites / SITES_PER_BLOCK);                // 16384 blocks
    kde_kernel<<<grid, KDE_THREADS, 0, stream>>>(x, out, tab);
}